// EmbedMatcher_59365037965913
// MI455X (gfx1250) — compile-verified
//
#include <hip/hip_runtime.h>
#include <hip/hip_bf16.h>
#include <math.h>
#include <stdint.h>

// ---------------- problem constants ----------------
#define V_SZ   200000
#define D_SZ   128
#define M_SZ   64
#define KNN_K  32
#define Q_SZ   4096
#define S_SZ   5
#define D2_SZ  256
#define D4_SZ  512
#define STEPS  4

typedef __attribute__((ext_vector_type(2))) float v2f;
typedef __attribute__((ext_vector_type(8))) float v8f;

// ---------------- helpers ----------------
__device__ __forceinline__ float wred32(float v) {
  #pragma unroll
  for (int off = 16; off > 0; off >>= 1) v += __shfl_xor(v, off, 32);
  return v;
}
__device__ __forceinline__ float sigmoidf_(float x) { return 1.0f / (1.0f + __expf(-x)); }

// CDNA5 async DMA: global -> LDS, 16B per lane, tracked by ASYNCcnt.
// saddr form: 64-bit SGPR base + 32-bit unsigned VGPR byte offset.
__device__ __forceinline__ void async_copy_b128(uint32_t lds_addr, uint32_t voff,
                                                const float* base) {
  asm volatile("global_load_async_to_lds_b128 %0, %1, %2"
               :: "v"(lds_addr), "v"(voff), "s"(base) : "memory");
}

// =====================================================================
// Stage 1: per-row (2Q+2S rows) neighbor means.
//   cat_mean[row]  = [ mean(rel_top32), mean(ent_top32) ]   (256)
//   catk_mean[row] = [ emb[0],          mean(knn_ent)   ]   (256)
// One wave32 per row. Embedding table (102MB) is L2-resident (192MB L2).
// =====================================================================
#define NB_WPB 8
__global__ void __launch_bounds__(256)
neighbor_stage1(const int* __restrict__ query, const int* __restrict__ support,
                const int* __restrict__ qlc, const int* __restrict__ qrc,
                const int* __restrict__ slc, const int* __restrict__ src_,
                const int* __restrict__ knn_table, const float* __restrict__ emb,
                float* __restrict__ cat_mean, float* __restrict__ catk_mean)
{
  const int w    = threadIdx.x >> 5;
  const int lane = threadIdx.x & 31;
  const int wave = blockIdx.x * NB_WPB + w;
  const int total = 2 * Q_SZ + 2 * S_SZ;
  const bool active = wave < total;
  const int row = active ? wave : 0;

  // segment decode
  int eid; const int* conn;
  if (row < Q_SZ)               { eid = query[row * 2];                 conn = qlc  + (size_t)row * M_SZ * 2; }
  else if (row < 2 * Q_SZ)      { int q = row - Q_SZ;                   eid = query[q * 2 + 1]; conn = qrc + (size_t)q * M_SZ * 2; }
  else if (row < 2 * Q_SZ + S_SZ){ int s = row - 2 * Q_SZ;              eid = support[s * 2];   conn = slc + (size_t)s * M_SZ * 2; }
  else                          { int s = row - 2 * Q_SZ - S_SZ;        eid = support[s * 2 + 1]; conn = src_ + (size_t)s * M_SZ * 2; }

  const int* kt = knn_table + (size_t)eid * KNN_K;
  __builtin_prefetch(kt, 0, 3);   // global_prefetch_b8

  // center vector: lane holds 4 consecutive floats
  const float4 cv = ((const float4*)(emb + (size_t)eid * D_SZ))[lane];
  float cn2 = wred32(cv.x * cv.x + cv.y * cv.y + cv.z * cv.z + cv.w * cv.w);
  const float cn = fmaxf(sqrtf(cn2), 1e-8f);

  __shared__ float sims[NB_WPB][M_SZ];
  __shared__ unsigned char sel[NB_WPB][M_SZ];

  // pass 1: cosine sims vs. center for all 64 neighbors
  for (int j = 0; j < M_SZ; ++j) {
    const int2 pe = ((const int2*)conn)[j];
    const float4 ev = ((const float4*)(emb + (size_t)pe.y * D_SZ))[lane];
    float d = cv.x * ev.x + cv.y * ev.y + cv.z * ev.z + cv.w * ev.w;
    float n = ev.x * ev.x + ev.y * ev.y + ev.z * ev.z + ev.w * ev.w;
    d = wred32(d); n = wred32(n);
    const float sim = d / (cn * fmaxf(sqrtf(n), 1e-8f));
    if (lane == 0) sims[w][j] = sim;
  }
  __syncthreads();

  // rank-based top-32 of 64 (ties broken by smaller index, matching lax.top_k)
  for (int j = lane; j < M_SZ; j += 32) {
    const float sj = sims[w][j];
    int rank = 0;
    #pragma unroll 8
    for (int i = 0; i < M_SZ; ++i) {
      const float si = sims[w][i];
      rank += (si > sj) || (si == sj && i < j);
    }
    sel[w][j] = (rank < KNN_K) ? 1 : 0;
  }
  __syncthreads();

  // pass 2: accumulate selected rel/ent rows (branch is wave-uniform)
  float4 ra = {0,0,0,0}, ea = {0,0,0,0};
  for (int j = 0; j < M_SZ; ++j) {
    if (sel[w][j]) {
      const int2 pe = ((const int2*)conn)[j];
      const float4 rv = ((const float4*)(emb + (size_t)pe.x * D_SZ))[lane];
      const float4 ev = ((const float4*)(emb + (size_t)pe.y * D_SZ))[lane];
      ra.x += rv.x; ra.y += rv.y; ra.z += rv.z; ra.w += rv.w;
      ea.x += ev.x; ea.y += ev.y; ea.z += ev.z; ea.w += ev.w;
    }
  }

  // knn branch: top_k over exactly 32 -> permutation -> plain mean
  float4 ka = {0,0,0,0};
  for (int j = 0; j < KNN_K; ++j) {
    const float4 ev = ((const float4*)(emb + (size_t)kt[j] * D_SZ))[lane];
    ka.x += ev.x; ka.y += ev.y; ka.z += ev.z; ka.w += ev.w;
  }
  const float4 e0 = ((const float4*)emb)[lane];   // emb[0] (rel half of knn cat)

  if (active) {
    const float inv = 1.0f / 32.0f;
    float4* cm  = (float4*)(cat_mean  + (size_t)row * D2_SZ);
    float4* ckm = (float4*)(catk_mean + (size_t)row * D2_SZ);
    float4 o;
    o.x = ra.x*inv; o.y = ra.y*inv; o.z = ra.z*inv; o.w = ra.w*inv; cm[lane]      = o;
    o.x = ea.x*inv; o.y = ea.y*inv; o.z = ea.z*inv; o.w = ea.w*inv; cm[lane + 32] = o;
    ckm[lane] = e0;
    o.x = ka.x*inv; o.y = ka.y*inv; o.z = ka.z*inv; o.w = ka.w*inv; ckm[lane + 32] = o;
  }
}

// =====================================================================
// Generic fp32 WMMA GEMM:  Out(Mr x N) = act( X(Mr x K) @ W(N x ldw).T[:, :K] + bias )
// One wave32 per 16x16 tile; 8 waves/block share one 16-row A tile
// (requires (N/16) % 8 == 0 -> block never straddles an M-tile row).
// A tile is double-buffered through LDS via global_load_async_to_lds_b128
// (cooperative: 256 threads x 16B = one 16x64 fp32 chunk), synced with
// s_wait_asynccnt + barrier. W streams straight from L2.
// K must be a multiple of 64; X padded to 16-row multiple. act:0 none,1 tanh,2 relu
// =====================================================================
#define GEMM_KC 64
__global__ void __launch_bounds__(256)
gemm_f32_wmma(const float* __restrict__ X, const float* __restrict__ W,
              const float* __restrict__ bias, const float* __restrict__ bias2,
              float* __restrict__ Out, int Mr, int N, int K, int ldw, int act)
{
  __shared__ float sA[2][16 * GEMM_KC];   // 8KB double buffer

  const int tid  = threadIdx.x;
  const int wv   = tid >> 5;
  const int lane = tid & 31;
  const int ntn  = N >> 4;
  const int ntm  = (Mr + 15) >> 4;
  const int ntiles = ntm * ntn;
  const int t0 = blockIdx.x * 8;
  const int t  = t0 + wv;
  const int m0 = (t0 / ntn) << 4;          // block-uniform since ntn % 8 == 0
  const int n0 = (t % ntn) << 4;
  const bool valid = t < ntiles;

  // cooperative A-chunk loader mapping: thread -> (row r, 4-float group c4)
  const int r  = tid >> 4;                 // 0..15
  const int c4 = (tid & 15) << 2;          // 0,4,...,60
  const uint32_t lds0  = (uint32_t)(uintptr_t)(&sA[0][r * GEMM_KC + c4]);
  const uint32_t lds1  = (uint32_t)(uintptr_t)(&sA[1][r * GEMM_KC + c4]);
  const uint32_t gbyte = (uint32_t)(((size_t)(m0 + r) * K + c4) * sizeof(float));

  const int lh = lane >> 4;                // half-wave selects K pair {0,1} vs {2,3}
  const int lm = lane & 15;
  const float2* wrow = (const float2*)(W + (size_t)(n0 + lm) * ldw);

  v8f acc = {};
  const int nck = K / GEMM_KC;

  async_copy_b128(lds0, gbyte, X);         // prefetch chunk 0
  for (int ck = 0; ck < nck; ++ck) {
    if (ck + 1 < nck) {
      async_copy_b128((ck + 1) & 1 ? lds1 : lds0,
                      gbyte + (uint32_t)((ck + 1) * GEMM_KC * sizeof(float)), X);
      asm volatile("s_wait_asynccnt 0x1" ::: "memory");   // previous chunk landed
    } else {
      asm volatile("s_wait_asynccnt 0x0" ::: "memory");
    }
    __syncthreads();                       // all waves' slices visible

    const float2* sa = (const float2*)&sA[ck & 1][lm * GEMM_KC];
    const int kw0 = (ck * GEMM_KC) >> 1;   // float2 index base into W row
    #pragma unroll
    for (int st = 0; st < GEMM_KC / 4; ++st) {
      const float2 av = sa[2 * st + lh];
      const float2 bv = wrow[kw0 + 2 * st + lh];
      v2f a; a[0] = av.x; a[1] = av.y;
      v2f b; b[0] = bv.x; b[1] = bv.y;
      acc = __builtin_amdgcn_wmma_f32_16x16x4_f32(false, a, false, b, (short)0, acc,
                                                  false, false);
    }
    __syncthreads();                       // buffer safe to overwrite next iter
  }

  if (!valid) return;
  const int n = n0 + lm;
  float bn = 0.0f;
  if (bias)  bn += bias[n];
  if (bias2) bn += bias2[n];
  #pragma unroll
  for (int v = 0; v < 8; ++v) {
    const int m = m0 + v + (lh ? 8 : 0);
    if (m < Mr) {
      float x = acc[v] + bn;
      if (act == 1)      x = tanhf(x);
      else if (act == 2) x = fmaxf(x, 0.0f);
      Out[(size_t)m * N + n] = x;
    }
  }
}

// =====================================================================
// Gate + combine: alpha = sigmoid([str,knn] . gate_W + gate_b);
// out = (1-alpha)*str + alpha*knn, scattered into the concat layout
// X (Q+S, 256): query rows 0..Q-1, support rows Q..Q+S-1; left half / right half.
// =====================================================================
__global__ void __launch_bounds__(256)
gate_combine(const float* __restrict__ strb, const float* __restrict__ knnb,
             const float* __restrict__ gate_W, const float* __restrict__ gate_b,
             float* __restrict__ X)
{
  const int lane = threadIdx.x & 31;
  const int wave = blockIdx.x * (blockDim.x >> 5) + (threadIdx.x >> 5);
  const int total = 2 * Q_SZ + 2 * S_SZ;
  if (wave >= total) return;
  const int row = wave;

  const float4 sv = ((const float4*)(strb + (size_t)row * D_SZ))[lane];
  const float4 kv = ((const float4*)(knnb + (size_t)row * D_SZ))[lane];
  const float4 g0 = ((const float4*)gate_W)[lane];        // gate_W[0:128]
  const float4 g1 = ((const float4*)gate_W)[lane + 32];   // gate_W[128:256]
  float p = sv.x*g0.x + sv.y*g0.y + sv.z*g0.z + sv.w*g0.w
          + kv.x*g1.x + kv.y*g1.y + kv.z*g1.z + kv.w*g1.w;
  p = wred32(p) + gate_b[0];
  const float a = sigmoidf_(p);

  float4 o;
  o.x = (1.0f-a)*sv.x + a*kv.x; o.y = (1.0f-a)*sv.y + a*kv.y;
  o.z = (1.0f-a)*sv.z + a*kv.z; o.w = (1.0f-a)*sv.w + a*kv.w;

  // scatter into (Q+S, 256): [ql|qr] for queries, [sl|sr] for supports
  size_t dst_row; int half;
  if (row < Q_SZ)                { dst_row = row;                         half = 0; }
  else if (row < 2*Q_SZ)         { dst_row = row - Q_SZ;                  half = 1; }
  else if (row < 2*Q_SZ + S_SZ)  { dst_row = Q_SZ + (row - 2*Q_SZ);       half = 0; }
  else                           { dst_row = Q_SZ + (row - 2*Q_SZ - S_SZ);half = 1; }
  ((float4*)(X + dst_row * D2_SZ + half * D_SZ))[lane] = o;
}

// =====================================================================
// LayerNorm (ddof=1) with residual: Y = LN(Z + X) * g + b, rows of 256
// =====================================================================
__global__ void __launch_bounds__(256)
ln_residual(const float* __restrict__ Z, const float* __restrict__ X,
            const float* __restrict__ g, const float* __restrict__ b,
            float* __restrict__ Y, int rows)
{
  const int lane = threadIdx.x & 31;
  const int row  = blockIdx.x * (blockDim.x >> 5) + (threadIdx.x >> 5);
  if (row >= rows) return;
  const float* zr = Z + (size_t)row * D2_SZ;
  const float* xr = X + (size_t)row * D2_SZ;
  float v[8], s = 0.0f, s2 = 0.0f;
  #pragma unroll
  for (int i = 0; i < 8; ++i) {
    const int c = lane * 8 + i;
    v[i] = zr[c] + xr[c];
    s += v[i]; s2 += v[i] * v[i];
  }
  s = wred32(s); s2 = wred32(s2);
  const float mu  = s * (1.0f / 256.0f);
  const float var = fmaxf((s2 - 256.0f * mu * mu) * (1.0f / 255.0f), 0.0f);
  const float inv = 1.0f / (sqrtf(var) + 0.001f);
  #pragma unroll
  for (int i = 0; i < 8; ++i) {
    const int c = lane * 8 + i;
    Y[(size_t)row * D2_SZ + c] = (v[i] - mu) * inv * g[c] + b[c];
  }
}

// support_g = mean over S support rows of Y; sgn = l2norm(support_g). 1 wave.
__global__ void support_g_kernel(const float* __restrict__ Y,
                                 float* __restrict__ sg, float* __restrict__ sgn)
{
  const int lane = threadIdx.x & 31;
  float v[8], s2 = 0.0f;
  #pragma unroll
  for (int i = 0; i < 8; ++i) {
    const int c = lane * 8 + i;
    float acc = 0.0f;
    for (int s = 0; s < S_SZ; ++s) acc += Y[(size_t)(Q_SZ + s) * D2_SZ + c];
    v[i] = acc * (1.0f / S_SZ);
    s2 += v[i] * v[i];
  }
  s2 = wred32(s2);
  const float inv = 1.0f / fmaxf(sqrtf(s2), 1e-12f);
  #pragma unroll
  for (int i = 0; i < 8; ++i) {
    const int c = lane * 8 + i;
    sg[c] = v[i];
    sgn[c] = v[i] * inv;
  }
}

// cvec[n] = bhh[n] + Whh[n, 256:512] . support_g   (r = support_g since softmax of 1 col == 1)
__global__ void cvec_kernel(const float* __restrict__ Whh, const float* __restrict__ bhh,
                            const float* __restrict__ sg, float* __restrict__ cvec)
{
  const int n = blockIdx.x * blockDim.x + threadIdx.x;
  if (n >= 4 * D4_SZ) return;
  const float* wr = Whh + (size_t)n * D4_SZ + D2_SZ;
  float acc = 0.0f;
  for (int k = 0; k < D2_SZ; ++k) acc += wr[k] * sg[k];
  cvec[n] = bhh[n] + acc;
}

// LSTM cell elementwise: gates = qx + glin(+opt) + cvec; update c, h.
__global__ void __launch_bounds__(256)
lstm_cell(const float* __restrict__ qx, const float* __restrict__ glin,
          const float* __restrict__ cvec, const float* __restrict__ queryg,
          float* __restrict__ c_state, float* __restrict__ h_state, int step)
{
  const int idx = blockIdx.x * blockDim.x + threadIdx.x;
  if (idx >= Q_SZ * D4_SZ) return;
  const int q = idx >> 9, n = idx & 511;
  const size_t base = (size_t)q * (4 * D4_SZ);
  float gi = qx[base + n]             + cvec[n];
  float gf = qx[base + D4_SZ + n]     + cvec[D4_SZ + n];
  float gg = qx[base + 2 * D4_SZ + n] + cvec[2 * D4_SZ + n];
  float go = qx[base + 3 * D4_SZ + n] + cvec[3 * D4_SZ + n];
  if (glin) {
    gi += glin[base + n];             gf += glin[base + D4_SZ + n];
    gg += glin[base + 2 * D4_SZ + n]; go += glin[base + 3 * D4_SZ + n];
  }
  float c = (step == 0) ? 0.0f : c_state[(size_t)q * D4_SZ + n];
  c = sigmoidf_(gf) * c + sigmoidf_(gi) * tanhf(gg);
  c_state[(size_t)q * D4_SZ + n] = c;
  const float hcell = sigmoidf_(go) * tanhf(c);
  if (n < D2_SZ)
    h_state[(size_t)q * D2_SZ + n] = queryg[(size_t)q * D2_SZ + n] + hcell;
}

// out[q] = dot(h[q], sgn) / max(||h[q]||, 1e-12)   (sgn pre-normalized)
__global__ void __launch_bounds__(256)
final_dot(const float* __restrict__ h, const float* __restrict__ sgn,
          float* __restrict__ out)
{
  const int lane = threadIdx.x & 31;
  const int q = blockIdx.x * (blockDim.x >> 5) + (threadIdx.x >> 5);
  if (q >= Q_SZ) return;
  const float* hr = h + (size_t)q * D2_SZ;
  float d = 0.0f, n2 = 0.0f;
  #pragma unroll
  for (int i = 0; i < 8; ++i) {
    const int c = lane * 8 + i;
    const float hv = hr[c];
    d += hv * sgn[c]; n2 += hv * hv;
  }
  d = wred32(d); n2 = wred32(n2);
  if (lane == 0) out[q] = d / fmaxf(sqrtf(n2), 1e-12f);
}

// =====================================================================
// Host launcher
// =====================================================================
extern "C" void kernel_launch(void* const* d_in, const int* in_sizes, int n_in,
                              void* d_out, int out_size, void* d_ws, size_t ws_size,
                              hipStream_t stream) {
  (void)in_sizes; (void)n_in; (void)out_size; (void)ws_size;
  const int*   query    = (const int*)  d_in[0];
  const int*   support  = (const int*)  d_in[1];
  const int*   qlc      = (const int*)  d_in[2];
  const int*   qrc      = (const int*)  d_in[3];
  const int*   slc      = (const int*)  d_in[4];
  const int*   src_     = (const int*)  d_in[5];
  const int*   knn_tab  = (const int*)  d_in[6];
  const float* emb      = (const float*)d_in[7];
  const float* gcn_W    = (const float*)d_in[8];
  const float* gcn_wb   = (const float*)d_in[9];
  const float* gcn_b    = (const float*)d_in[10];
  const float* gate_W   = (const float*)d_in[11];
  const float* gate_b   = (const float*)d_in[12];
  const float* se_W1    = (const float*)d_in[13];
  const float* se_b1    = (const float*)d_in[14];
  const float* se_W2    = (const float*)d_in[15];
  const float* se_b2    = (const float*)d_in[16];
  const float* ln_g     = (const float*)d_in[17];
  const float* ln_b     = (const float*)d_in[18];
  const float* Wih      = (const float*)d_in[19];
  const float* Whh      = (const float*)d_in[20];
  const float* bih      = (const float*)d_in[21];
  const float* bhh      = (const float*)d_in[22];
  float* out = (float*)d_out;

  // workspace layout (floats); padded row counts for 16-row WMMA tiles
  const int NB   = 2 * Q_SZ + 2 * S_SZ;  // 8202
  const int NBp  = (NB + 15) & ~15;      // 8208
  const int QS   = Q_SZ + S_SZ;          // 4101
  const int QSp  = (QS + 15) & ~15;      // 4112
  float* ws = (float*)d_ws;
  size_t off = 0;
  auto alloc = [&](size_t n) { float* p = ws + off; off += n; return p; };
  float* cat_mean  = alloc((size_t)NBp * D2_SZ);
  float* catk_mean = alloc((size_t)NBp * D2_SZ);
  float* strb      = alloc((size_t)NBp * D_SZ);
  float* knnb      = alloc((size_t)NBp * D_SZ);
  float* X         = alloc((size_t)QSp * D2_SZ);
  float* h1        = alloc((size_t)QSp * D4_SZ);
  float* z         = alloc((size_t)QSp * D2_SZ);
  float* Y         = alloc((size_t)QSp * D2_SZ);
  float* sg        = alloc(D2_SZ);
  float* sgn       = alloc(D2_SZ);
  float* cvec      = alloc(4 * D4_SZ);
  float* qx        = alloc((size_t)Q_SZ * 4 * D4_SZ);
  float* glin      = alloc((size_t)Q_SZ * 4 * D4_SZ);
  float* cst       = alloc((size_t)Q_SZ * D4_SZ);
  float* hst       = alloc((size_t)Q_SZ * D2_SZ);

  auto gemm = [&](const float* Xp, const float* Wp, const float* b1p,
                  const float* b2p, float* O, int Mr, int N, int K, int ldw, int act) {
    const int tiles = (((Mr + 15) >> 4)) * (N >> 4);
    gemm_f32_wmma<<<(tiles + 7) / 8, 256, 0, stream>>>(Xp, Wp, b1p, b2p, O,
                                                       Mr, N, K, ldw, act);
  };

  // ---- stage 1: neighbor means ----
  neighbor_stage1<<<(NB + NB_WPB - 1) / NB_WPB, 256, 0, stream>>>(
      query, support, qlc, qrc, slc, src_, knn_tab, emb, cat_mean, catk_mean);

  // ---- stage 2: structural / knn GEMM (tanh) + gate combine ----
  gemm(cat_mean,  gcn_W, gcn_wb, gcn_b, strb, NB, D_SZ, D2_SZ, D2_SZ, /*tanh*/1);
  gemm(catk_mean, gcn_W, gcn_wb, gcn_b, knnb, NB, D_SZ, D2_SZ, D2_SZ, /*tanh*/1);
  gate_combine<<<(NB + 7) / 8, 256, 0, stream>>>(strb, knnb, gate_W, gate_b, X);

  // ---- stage 3: support encoder (shared for query+support rows) ----
  gemm(X,  se_W1, se_b1, nullptr, h1, QS, D4_SZ, D2_SZ, D2_SZ, /*relu*/2);
  gemm(h1, se_W2, se_b2, nullptr, z,  QS, D2_SZ, D4_SZ, D4_SZ, 0);
  ln_residual<<<(QS + 7) / 8, 256, 0, stream>>>(z, X, ln_g, ln_b, Y, QS);
  support_g_kernel<<<1, 32, 0, stream>>>(Y, sg, sgn);

  // ---- stage 4: LSTM query encoder ----
  // qx = query_g @ Wih.T + bih  (query_g = first Q rows of Y)
  gemm(Y, Wih, bih, nullptr, qx, Q_SZ, 4 * D4_SZ, D2_SZ, D2_SZ, 0);
  // constant gate term: bhh + Whh[:,256:] @ support_g (attn over 1 col == 1)
  cvec_kernel<<<(4 * D4_SZ + 255) / 256, 256, 0, stream>>>(Whh, bhh, sg, cvec);

  const int cellBlocks = (Q_SZ * D4_SZ + 255) / 256;
  for (int step = 0; step < STEPS; ++step) {
    const float* gl = nullptr;
    const float* cv = bhh;   // step 0: h_r == 0 -> gates = qx + bhh
    if (step > 0) {
      gemm(hst, Whh, nullptr, nullptr, glin, Q_SZ, 4 * D4_SZ, D2_SZ, D4_SZ, 0);
      gl = glin; cv = cvec;
    }
    lstm_cell<<<cellBlocks, 256, 0, stream>>>(qx, gl, cv, Y, cst, hst, step);
  }

  // ---- stage 5: normalized dot with support_g ----
  final_dot<<<(Q_SZ + 7) / 8, 256, 0, stream>>>(hst, sgn, out);
}